// InterIntraLoss_18640158065097
// MI455X (gfx1250) — compile-verified
//
#include <hip/hip_runtime.h>
#include <math.h>
#include <cstddef>

typedef float v2f __attribute__((ext_vector_type(2)));
typedef float v8f __attribute__((ext_vector_type(8)));

#define NCLS 16
#define DDIM 1024
#define MARGIN_ 1.0f

struct Ws {
  float ce_sum;
  float pad0[3];
  int   counts[NCLS];
  float dist_sums[NCLS];
  float sums[NCLS * DDIM];        // 16-byte aligned (offset 144)
  float centroids[NCLS * DDIM];   // offset 65680, 16-byte aligned
};

// ---------------------------------------------------------------- zero ws
__global__ void init_ws_kernel(int* __restrict__ w, int ndwords) {
  int i = blockIdx.x * blockDim.x + threadIdx.x;
  if (i < ndwords) w[i] = 0;
}

// ------------------------------------------- cross-entropy + class counts
__global__ void ce_counts_kernel(const float* __restrict__ logits,
                                 const int* __restrict__ labels,
                                 Ws* __restrict__ ws) {
  __shared__ float sce;
  __shared__ int scnt[NCLS];
  if (threadIdx.x == 0) sce = 0.f;
  if (threadIdx.x < NCLS) scnt[threadIdx.x] = 0;
  __syncthreads();

  int i = blockIdx.x * blockDim.x + threadIdx.x;
  const float4* row = (const float4*)(logits + (size_t)i * NCLS);
  float4 q[4];
  float m = -INFINITY;
#pragma unroll
  for (int j = 0; j < 4; ++j) {
    q[j] = row[j];
    m = fmaxf(m, fmaxf(fmaxf(q[j].x, q[j].y), fmaxf(q[j].z, q[j].w)));
  }
  float s = 0.f;
#pragma unroll
  for (int j = 0; j < 4; ++j)
    s += expf(q[j].x - m) + expf(q[j].y - m) + expf(q[j].z - m) + expf(q[j].w - m);

  int lab = labels[i];
  float xl = logits[(size_t)i * NCLS + lab];   // cached reload; avoids scratch
  float nll = -(xl - m - logf(s));

  atomicAdd(&sce, nll);           // ds_add_f32
  atomicAdd(&scnt[lab], 1);
  __syncthreads();
  if (threadIdx.x == 0) atomicAdd(&ws->ce_sum, sce);
  if (threadIdx.x < NCLS) atomicAdd(&ws->counts[threadIdx.x], scnt[threadIdx.x]);
}

// -------- per-class feature sums as onehot^T @ features via fp32 WMMA ----
// Each wave: one 16-wide D tile x 512 batch rows, K=4 per WMMA (128 WMMAs).
// Uniform scalar trip counter -> s_cbranch loop, EXEC stays all-1s (WMMA req).
// 8 waves of a block share the D tile: reduce tiles in LDS, then one global
// atomic per element per block (8x fewer device-scope atomics).
#define ROWS_PER_WAVE 512
__global__ void seg_sum_wmma_kernel(const float* __restrict__ features,
                                    const int* __restrict__ labels,
                                    float* __restrict__ sums) {
  const int lane  = threadIdx.x & 31;
  const int wave  = threadIdx.x >> 5;                 // 0..7
  const int d0    = blockIdx.x * 16;                  // D tile
  const int chunk = blockIdx.y * 8 + wave;            // batch chunk
  const int r0    = chunk * ROWS_PER_WAVE;

  const int n     = lane & 15;                        // N col (also M for A)
  const int khalf = (lane >> 4) * 2;                  // K half-select 0 / 2

  const float* fptr = features + (size_t)(r0 + khalf) * DDIM + d0 + n;
  const int*   lptr = labels + r0 + khalf;

  v8f acc = {};
#pragma unroll 4
  for (int it = 0; it < ROWS_PER_WAVE / 4; ++it) {
    v2f b;                                            // B[4x16] tile of features
    b.x = fptr[0];
    b.y = fptr[DDIM];
    const int l0 = lptr[0];
    const int l1 = lptr[1];
    v2f a;                                            // A[16x4] one-hot (exact fp32)
    a.x = (l0 == n) ? 1.0f : 0.0f;
    a.y = (l1 == n) ? 1.0f : 0.0f;
    acc = __builtin_amdgcn_wmma_f32_16x16x4_f32(false, a, false, b,
                                                (short)0, acc, false, false);
    fptr += 4 * DDIM;
    lptr += 4;
  }

  // Cross-wave reduction of the 16x16 tile in LDS (ds_add_f32).
  __shared__ float stile[NCLS * NCLS];                // blockDim.x == 256
  stile[threadIdx.x] = 0.f;
  __syncthreads();
  const int M0 = (lane < 16) ? 0 : 8;                 // C/D layout rows
#pragma unroll
  for (int r = 0; r < 8; ++r)
    atomicAdd(&stile[(M0 + r) * NCLS + n], acc[r]);
  __syncthreads();
  const int em = threadIdx.x >> 4;                    // element row
  const int en = threadIdx.x & 15;                    // element col
  atomicAdd(&sums[(size_t)em * DDIM + d0 + en], stile[threadIdx.x]);
}

// ------------------------------------------------- centroids = sums/count
__global__ void centroid_kernel(Ws* __restrict__ ws) {
  int i = blockIdx.x * blockDim.x + threadIdx.x;      // 16*1024
  int c = i >> 10;
  int cnt = ws->counts[c];
  float denom = (cnt > 0) ? (float)cnt : 1.0f;
  ws->centroids[i] = ws->sums[i] / denom;
}

// ------------- intra pass: per-row distance to centroid, segment-summed --
__global__ void intra_kernel(const float* __restrict__ features,
                             const int* __restrict__ labels,
                             Ws* __restrict__ ws) {
  __shared__ float sdist[NCLS];
  if (threadIdx.x < NCLS) sdist[threadIdx.x] = 0.f;
  __syncthreads();

  const int wave = threadIdx.x >> 5;
  const int lane = threadIdx.x & 31;
  const int row  = blockIdx.x * 8 + wave;
  const int lab  = labels[row];
  const float* f = features + (size_t)row * DDIM;
  const float* c = ws->centroids + (size_t)lab * DDIM;

  float acc = 0.f;
#pragma unroll
  for (int d = lane * 4; d < DDIM; d += 128) {
    float4 fv = *(const float4*)(f + d);
    float4 cv = *(const float4*)(c + d);
    float dx = fv.x - cv.x, dy = fv.y - cv.y, dz = fv.z - cv.z, dw = fv.w - cv.w;
    acc += dx * dx + dy * dy + dz * dz + dw * dw;
  }
#pragma unroll
  for (int off = 16; off > 0; off >>= 1)
    acc += __shfl_xor(acc, off);
  if (lane == 0) atomicAdd(&sdist[lab], sqrtf(acc));
  __syncthreads();
  if (threadIdx.x < NCLS) atomicAdd(&ws->dist_sums[threadIdx.x], sdist[threadIdx.x]);
}

// -------- finale: centroid Gram via fp32 WMMA (one wave) + combine -------
__global__ void final_kernel(Ws* __restrict__ ws, float* __restrict__ out, int B) {
  const int lane = threadIdx.x;                       // 0..31
  const float* cent = ws->centroids;

  // Gram G = C * C^T.  A(16x4) and B(4x16=C^T tile) have identical per-lane
  // register contents for this shape, so load once and feed both operands.
  const int m    = lane & 15;
  const int koff = (lane >> 4) * 2;
  const float* cptr = cent + (size_t)m * DDIM + koff;
  v8f g = {};
#pragma unroll 4
  for (int it = 0; it < DDIM / 4; ++it) {
    v2f a;
    a.x = cptr[0];
    a.y = cptr[1];
    g = __builtin_amdgcn_wmma_f32_16x16x4_f32(false, a, false, a,
                                              (short)0, g, false, false);
    cptr += 4;
  }
  __shared__ float G[NCLS][NCLS];
  const int M0 = (lane < 16) ? 0 : 8;
#pragma unroll
  for (int r = 0; r < 8; ++r) G[M0 + r][m] = g[r];
  __syncthreads();

  if (lane == 0) {
    bool valid[NCLS];
    float n_valid = 0.f, intra_sum = 0.f;
#pragma unroll
    for (int c = 0; c < NCLS; ++c) {
      int cnt = ws->counts[c];
      valid[c] = cnt > 0;
      if (valid[c]) { n_valid += 1.f; intra_sum += ws->dist_sums[c] / (float)cnt; }
    }
    float intra = (n_valid > 0.f) ? intra_sum / n_valid : 0.f;
    float ce = ws->ce_sum / (float)B;

    float inter_sum = 0.f, n_pairs = 0.f;
    for (int i = 0; i < NCLS; ++i)
      for (int j = i + 1; j < NCLS; ++j)
        if (valid[i] && valid[j]) {
          float d2 = G[i][i] + G[j][j] - 2.f * G[i][j];
          float d  = sqrtf(fmaxf(d2, 0.f));
          float t  = MARGIN_ - d;
          if (t > 0.f) inter_sum += t;
          n_pairs += 1.f;
        }
    float inter = (n_pairs > 0.f) ? inter_sum / n_pairs : 0.f;
    out[0] = ce + inter + intra;   // ALPHA=BETA=GAMMA=1
  }
}

// ------------------------------------------------------------------ launch
extern "C" void kernel_launch(void* const* d_in, const int* in_sizes, int n_in,
                              void* d_out, int out_size, void* d_ws, size_t ws_size,
                              hipStream_t stream) {
  const float* logits   = (const float*)d_in[0];
  const int*   labels   = (const int*)d_in[1];
  const float* features = (const float*)d_in[2];
  float* out = (float*)d_out;
  Ws* ws = (Ws*)d_ws;
  const int B = in_sizes[1];                          // 65536

  const int zero_dwords = (int)(offsetof(Ws, centroids) / 4);
  init_ws_kernel<<<(zero_dwords + 255) / 256, 256, 0, stream>>>((int*)d_ws, zero_dwords);

  ce_counts_kernel<<<B / 256, 256, 0, stream>>>(logits, labels, ws);

  dim3 gseg(DDIM / 16, B / (8 * ROWS_PER_WAVE));      // 64 x 16 blocks, 8 waves each
  seg_sum_wmma_kernel<<<gseg, 256, 0, stream>>>(features, labels, ws->sums);

  centroid_kernel<<<(NCLS * DDIM) / 256, 256, 0, stream>>>(ws);

  intra_kernel<<<B / 8, 256, 0, stream>>>(features, labels, ws);

  final_kernel<<<1, 32, 0, stream>>>(ws, out, B);
}